// GNNTransductiveEdgeHead_80659485819067
// MI455X (gfx1250) — compile-verified
//
#include <hip/hip_runtime.h>
#include <hip/hip_bf16.h>

typedef float v2f __attribute__((ext_vector_type(2)));
typedef float v8f __attribute__((ext_vector_type(8)));

#define DIM 128
#define MTILE 64          // rows per block (4 x 16-row WMMA tiles per wave)

// ---------------------------------------------------------------------------
// Fused 2-layer MLP: h2 = relu(relu(x@W1 + b1)@W2 + b2)
// Block = 256 threads = 8 waves (wave32). Each block computes 64 rows of h2.
// Wave w owns output columns [16w, 16w+16) for all 4 row-tiles:
//   - one B (weight) load feeds 4 independent WMMAs (4 accumulators)
//   - W1/W2 L2 traffic amortized 4x vs a 16-row tile
// Layer-1 result tile (64 x 128, 32 KB) is staged in LDS.
// WMMA: V_WMMA_F32_16X16X4_F32 (full f32 precision, matches reference math).
// ---------------------------------------------------------------------------
__global__ __launch_bounds__(256) void mlp2_fused_kernel(
    const float* __restrict__ x,
    const float* __restrict__ W1, const float* __restrict__ b1,
    const float* __restrict__ W2, const float* __restrict__ b2,
    float* __restrict__ h2, int n_nodes)
{
    __shared__ float h1s[MTILE * DIM];       // 32 KB layer-1 tile

    const int wave  = threadIdx.x >> 5;      // 0..7 -> column tile
    const int lane  = threadIdx.x & 31;
    const int l15   = lane & 15;             // M index (A) / N index (B,C,D)
    const int khalf = lane >> 4;             // 0: K={0,1}, 1: K={2,3}
    const int n0    = wave << 4;             // this wave's column base
    const int m0    = blockIdx.x * MTILE;    // this block's row base

    // clamped source rows for A loads (keeps EXEC uniform; stores are guarded)
    const float* __restrict__ xrow[4];
#pragma unroll
    for (int rt = 0; rt < 4; ++rt) {
        int arow = m0 + rt * 16 + l15;
        if (arow > n_nodes - 1) arow = n_nodes - 1;
        xrow[rt] = x + (size_t)arow * DIM;
    }

    // ------------------ Layer 1: c[rt] = x_tile[rt] @ W1 ------------------
    v8f c0 = {}, c1 = {}, c2 = {}, c3 = {};
    for (int k0 = 0; k0 < DIM; k0 += 4) {
        // B: 4x16 f32 tile of W1 (row-major [K][N]). lane=N; K split by lane half.
        v2f b;
        b.x = W1[(size_t)(k0 + 2 * khalf + 0) * DIM + n0 + l15];
        b.y = W1[(size_t)(k0 + 2 * khalf + 1) * DIM + n0 + l15];
        // A: 16x4 f32 tiles of x. lane=M; VGPR pair holds K = k0+2*khalf, +1
        v2f a0 = *(const v2f*)(xrow[0] + k0 + 2 * khalf);
        v2f a1 = *(const v2f*)(xrow[1] + k0 + 2 * khalf);
        v2f a2 = *(const v2f*)(xrow[2] + k0 + 2 * khalf);
        v2f a3 = *(const v2f*)(xrow[3] + k0 + 2 * khalf);
        c0 = __builtin_amdgcn_wmma_f32_16x16x4_f32(false, a0, false, b, (short)0, c0, false, false);
        c1 = __builtin_amdgcn_wmma_f32_16x16x4_f32(false, a1, false, b, (short)0, c1, false, false);
        c2 = __builtin_amdgcn_wmma_f32_16x16x4_f32(false, a2, false, b, (short)0, c2, false, false);
        c3 = __builtin_amdgcn_wmma_f32_16x16x4_f32(false, a3, false, b, (short)0, c3, false, false);
    }

    // bias + ReLU, stage into LDS. D element (vgpr r, lane) = D[r+8*khalf][l15]
    {
        const float bias = b1[n0 + l15];
        v8f acc[4] = { c0, c1, c2, c3 };
#pragma unroll
        for (int rt = 0; rt < 4; ++rt) {
#pragma unroll
            for (int r = 0; r < 8; ++r) {
                const int m = rt * 16 + r + 8 * khalf;
                float v = acc[rt][r] + bias;
                v = v > 0.0f ? v : 0.0f;
                h1s[m * DIM + n0 + l15] = v;
            }
        }
    }
    __syncthreads();

    // ------------------ Layer 2: d[rt] = h1_tile[rt] @ W2 ------------------
    v8f d0 = {}, d1 = {}, d2 = {}, d3 = {};
    for (int k0 = 0; k0 < DIM; k0 += 4) {
        v2f b;
        b.x = W2[(size_t)(k0 + 2 * khalf + 0) * DIM + n0 + l15];
        b.y = W2[(size_t)(k0 + 2 * khalf + 1) * DIM + n0 + l15];
        v2f a0 = *(const v2f*)(h1s + (0 * 16 + l15) * DIM + k0 + 2 * khalf);  // ds_load_b64
        v2f a1 = *(const v2f*)(h1s + (1 * 16 + l15) * DIM + k0 + 2 * khalf);
        v2f a2 = *(const v2f*)(h1s + (2 * 16 + l15) * DIM + k0 + 2 * khalf);
        v2f a3 = *(const v2f*)(h1s + (3 * 16 + l15) * DIM + k0 + 2 * khalf);
        d0 = __builtin_amdgcn_wmma_f32_16x16x4_f32(false, a0, false, b, (short)0, d0, false, false);
        d1 = __builtin_amdgcn_wmma_f32_16x16x4_f32(false, a1, false, b, (short)0, d1, false, false);
        d2 = __builtin_amdgcn_wmma_f32_16x16x4_f32(false, a2, false, b, (short)0, d2, false, false);
        d3 = __builtin_amdgcn_wmma_f32_16x16x4_f32(false, a3, false, b, (short)0, d3, false, false);
    }

    // bias + ReLU, store h2 tiles to global (L2-resident workspace)
    {
        const float bias = b2[n0 + l15];
        v8f acc[4] = { d0, d1, d2, d3 };
#pragma unroll
        for (int rt = 0; rt < 4; ++rt) {
#pragma unroll
            for (int r = 0; r < 8; ++r) {
                const int m = m0 + rt * 16 + r + 8 * khalf;
                if (m < n_nodes) {
                    float v = acc[rt][r] + bias;
                    v = v > 0.0f ? v : 0.0f;
                    h2[(size_t)m * DIM + n0 + l15] = v;
                }
            }
        }
    }
}

// ---------------------------------------------------------------------------
// Edge decode: pred[e] = dot(h2[src[e]], h2[dst[e]]); also emit labels as f32.
// One wave (32 lanes) per edge; each lane handles a float4 chunk of the
// 128-wide rows (512B coalesced reads, served from L2 since h2 fits in L2).
// ---------------------------------------------------------------------------
__global__ __launch_bounds__(256) void edge_dot_kernel(
    const float* __restrict__ h2,
    const long long* __restrict__ edge_index,
    const long long* __restrict__ edge_label,
    float* __restrict__ out, int n_edges)
{
    const int e    = (blockIdx.x * blockDim.x + threadIdx.x) >> 5;
    const int lane = threadIdx.x & 31;
    if (e >= n_edges) return;

    const long long s = edge_index[e];
    const long long t = edge_index[(size_t)n_edges + e];

    const float4 av = ((const float4*)(h2 + (size_t)s * DIM))[lane];
    const float4 bv = ((const float4*)(h2 + (size_t)t * DIM))[lane];

    float sum = av.x * bv.x + av.y * bv.y + av.z * bv.z + av.w * bv.w;

    // wave32 butterfly reduction
#pragma unroll
    for (int off = 16; off > 0; off >>= 1)
        sum += __shfl_xor(sum, off, 32);

    if (lane == 0) {
        out[e] = sum;                                    // pred
        out[(size_t)n_edges + e] = (float)edge_label[e]; // label (tuple elem 2)
    }
}

// ---------------------------------------------------------------------------
extern "C" void kernel_launch(void* const* d_in, const int* in_sizes, int n_in,
                              void* d_out, int out_size, void* d_ws, size_t ws_size,
                              hipStream_t stream) {
    const float*     x          = (const float*)d_in[0];
    const long long* edge_index = (const long long*)d_in[1];
    const long long* edge_label = (const long long*)d_in[2];
    const float*     W1         = (const float*)d_in[3];
    const float*     b1         = (const float*)d_in[4];
    const float*     W2         = (const float*)d_in[5];
    const float*     b2         = (const float*)d_in[6];

    const int n_nodes = in_sizes[0] / DIM;
    const int n_edges = in_sizes[2];

    float* h2  = (float*)d_ws;                 // n_nodes * 128 * 4 = 51.2 MB
    float* out = (float*)d_out;

    // MLP: one 64-row tile per block, 8 waves of 16 columns x 4 row-tiles
    const int mlp_blocks = (n_nodes + MTILE - 1) / MTILE;
    mlp2_fused_kernel<<<mlp_blocks, 256, 0, stream>>>(x, W1, b1, W2, b2, h2, n_nodes);

    // Edge decode: one wave per edge, 8 waves per block
    const int edge_blocks = (n_edges + 7) / 8;
    edge_dot_kernel<<<edge_blocks, 256, 0, stream>>>(h2, edge_index, edge_label,
                                                     out, n_edges);
}